// QVTree_18588618457675
// MI455X (gfx1250) — compile-verified
//
#include <hip/hip_runtime.h>

typedef __attribute__((ext_vector_type(2))) float v2f;
typedef __attribute__((ext_vector_type(8))) float v8f;

#define BATCH   64
#define NTOK    4096
#define DIM     256
#define GRID_H  64
#define M_NODES 5461
#define LEAF_OFF 1365

// level offsets of the full quadtree (BFS / Morton order)
__device__ const int kOFF[8] = {0, 1, 5, 21, 85, 341, 1365, 5461};

__device__ __forceinline__ unsigned morton2(unsigned r, unsigned c) {
  unsigned z = 0;
#pragma unroll
  for (int i = 0; i < 6; ++i)
    z |= (((c >> i) & 1u) << (2 * i)) | (((r >> i) & 1u) << (2 * i + 1));
  return z;
}

__device__ __forceinline__ void demorton2(unsigned z, unsigned& r, unsigned& c) {
  r = 0; c = 0;
#pragma unroll
  for (int i = 0; i < 6; ++i) {
    c |= ((z >> (2 * i)) & 1u) << i;
    r |= ((z >> (2 * i + 1)) & 1u) << i;
  }
}

// -------------------------------------------------------------------------
// Kernel 1: y[b,n] = dot(x[b,n,:], q[b,:])  via V_WMMA_F32_16X16X4_F32.
// One wave per 16-token tile. B = q broadcast into all 16 columns, so every
// output column equals y; column N=0 lives in lanes 0 (M=0..7) / 16 (M=8..15).
//
// K-remap for wide loads: per iteration t each lane does ONE b128 load of
// x[row, 8t + 4*(lane>=16) .. +3] (and the matching q quad). Elements {0,1}
// feed WMMA #1 (K={0,1} lanes 0-15 = cols 8t,8t+1; K={2,3} lanes 16-31 =
// cols 8t+4,8t+5), elements {2,3} feed WMMA #2 (cols 8t+2,8t+3 / 8t+6,8t+7).
// Every dim counted exactly once; A and B use identical K mappings.
// -------------------------------------------------------------------------
__global__ __launch_bounds__(32) void qvt_gemv_wmma(const float* __restrict__ x,
                                                    const float* __restrict__ q,
                                                    float* __restrict__ y) {
  const int lane = threadIdx.x;            // 0..31
  const int tile = blockIdx.x;             // 0..255
  const int b    = blockIdx.y;             // 0..63
  const int row  = tile * 16 + (lane & 15);
  const int sel4 = (lane >> 4) << 2;       // 0 or 4

  const float* xrow = x + ((size_t)b * NTOK + row) * DIM + sel4;
  const float* qrow = q + (size_t)b * DIM + sel4;

  v8f acc = {0.f, 0.f, 0.f, 0.f, 0.f, 0.f, 0.f, 0.f};
#pragma unroll 2
  for (int t = 0; t < DIM / 8; ++t) {
    const float4 av = *(const float4*)(xrow + 8 * t);
    const float4 qv = *(const float4*)(qrow + 8 * t);
    v2f a0, b0, a1, b1;
    a0[0] = av.x; a0[1] = av.y;  b0[0] = qv.x; b0[1] = qv.y;
    a1[0] = av.z; a1[1] = av.w;  b1[0] = qv.z; b1[1] = qv.w;
    acc = __builtin_amdgcn_wmma_f32_16x16x4_f32(false, a0, false, b0,
                                                (short)0, acc, false, false);
    acc = __builtin_amdgcn_wmma_f32_16x16x4_f32(false, a1, false, b1,
                                                (short)0, acc, false, false);
  }

  if ((lane & 15) == 0) {                  // lanes 0 and 16 hold column N=0
    union { v8f v; float f[8]; } u; u.v = acc;
    const int mbase = tile * 16 + ((lane >> 4) * 8);
    float* yo = y + (size_t)b * NTOK + mbase;
#pragma unroll
    for (int j = 0; j < 8; ++j) yo[j] = u.f[j];
  }
}

// -------------------------------------------------------------------------
// Kernel 2: per-batch scalar integral image + quadtree descent + compaction.
// One 256-thread block per batch. ~51 KB static LDS.
// -------------------------------------------------------------------------
__global__ __launch_bounds__(256) void qvt_tree(const float* __restrict__ y,
                                                int* __restrict__ perm,
                                                int* __restrict__ Karr) {
  __shared__ float         Ps[65 * 65];     // padded 2D prefix of y
  __shared__ float         ss[M_NODES];     // node scores
  __shared__ unsigned char vis[M_NODES];
  __shared__ unsigned char cov[M_NODES];
  __shared__ int           cnts[256];

  const int tid = threadIdx.x;
  const int b   = blockIdx.x;
  const float* yb = y + (size_t)b * NTOK;

  // ---- 2D inclusive prefix with zero pad row/col ----
  if (tid < 65) { Ps[tid] = 0.f; Ps[tid * 65] = 0.f; }
  __syncthreads();
  if (tid < 64) {                           // row cumsum, one row per thread
    float run = 0.f;
    const float* yr = yb + tid * 64;
    for (int c = 0; c < 64; ++c) { run += yr[c]; Ps[(tid + 1) * 65 + (c + 1)] = run; }
  }
  __syncthreads();
  if (tid < 64) {                           // column cumsum in place
    const int c = tid + 1;
    float run = 0.f;
    for (int r = 1; r <= 64; ++r) { run += Ps[r * 65 + c]; Ps[r * 65 + c] = run; }
  }
  __syncthreads();

  // ---- node scores, level by level (Morton-regular tree) ----
  for (int l = 0; l <= 6; ++l) {
    const int cnt  = 1 << (2 * l);
    const int base = kOFF[l];
    const int size = GRID_H >> l;
    const float inv = 1.0f / (float)(size * size);
    for (int i = tid; i < cnt; i += 256) {
      unsigned r, c; demorton2((unsigned)i, r, c);
      const int r0 = r * size, c0 = c * size, r1 = r0 + size, c1 = c0 + size;
      const float sum = Ps[r1 * 65 + c1] - Ps[r0 * 65 + c1]
                      - Ps[r1 * 65 + c0] + Ps[r0 * 65 + c0];
      ss[base + i] = sum * inv;
    }
  }
  __syncthreads();

  // ---- visited: descend only while score improves ----
  if (tid == 0) vis[0] = 1;
  __syncthreads();
  for (int l = 1; l <= 6; ++l) {
    const int cnt = 1 << (2 * l), base = kOFF[l], pbase = kOFF[l - 1];
    for (int i = tid; i < cnt; i += 256) {
      const int m = base + i, p = pbase + (i >> 2);
      vis[m] = (unsigned char)(vis[p] && (ss[m] > ss[p]));
    }
    __syncthreads();
  }

  // ---- selected (visited & no better child), propagated down as covered ----
  if (tid == 0) {
    const float sm = ss[0];
    const bool any = (ss[1] > sm) || (ss[2] > sm) || (ss[3] > sm) || (ss[4] > sm);
    cov[0] = (unsigned char)(vis[0] && !any);
  }
  __syncthreads();
  for (int l = 1; l <= 6; ++l) {
    const int cnt = 1 << (2 * l), base = kOFF[l], pbase = kOFF[l - 1];
    for (int i = tid; i < cnt; i += 256) {
      const int m = base + i;
      unsigned char cv = cov[pbase + (i >> 2)];
      if (!cv && vis[m]) {
        bool any = false;
        if (l < 6) {
          const int cb = kOFF[l + 1] + 4 * i;
          const float sm = ss[m];
          any = (ss[cb] > sm) || (ss[cb + 1] > sm) ||
                (ss[cb + 2] > sm) || (ss[cb + 3] > sm);
        }
        cv = (unsigned char)(!any);
      }
      cov[m] = cv;
    }
    __syncthreads();
  }

  // ---- stable compaction: covered tokens first, by token index ----
  const int nbase = tid * 16;
  unsigned char mk[16];
  int myCnt = 0;
#pragma unroll
  for (int j = 0; j < 16; ++j) {
    const int n = nbase + j;
    const unsigned r = (unsigned)(n >> 6), c = (unsigned)(n & 63);
    mk[j] = cov[LEAF_OFF + morton2(r, c)];
    myCnt += mk[j];
  }
  cnts[tid] = myCnt;
  __syncthreads();
  if (tid == 0) {                           // tiny serial exclusive scan
    int tot = 0;
    for (int i = 0; i < 256; ++i) { const int t = cnts[i]; cnts[i] = tot; tot += t; }
    Karr[b] = tot;
  }
  __syncthreads();
  int pos = cnts[tid];
  int* pb = perm + (size_t)b * NTOK;
#pragma unroll
  for (int j = 0; j < 16; ++j)
    if (mk[j]) pb[pos++] = nbase + j;
}

// -------------------------------------------------------------------------
// Kernel 3: write feats_pad (gathered rows / zeros) and mask_sorted.
// 256-thread block handles 4 output rows (64 lanes x float4 per row).
// -------------------------------------------------------------------------
__global__ __launch_bounds__(256) void qvt_write(const float* __restrict__ x,
                                                 const int* __restrict__ perm,
                                                 const int* __restrict__ Karr,
                                                 float* __restrict__ feats,
                                                 float* __restrict__ maskOut) {
  const int sub = threadIdx.x >> 6;        // 0..3 : row within block
  const int ln  = threadIdx.x & 63;        // 0..63: float4 slot within row
  const int p   = blockIdx.x * 4 + sub;
  const int b   = blockIdx.y;
  const int K   = Karr[b];

  float4* dst = (float4*)(feats + ((size_t)b * NTOK + p) * DIM);
  if (p < K) {
    const int n = perm[(size_t)b * NTOK + p];
    const float4* src = (const float4*)(x + ((size_t)b * NTOK + n) * DIM);
    dst[ln] = src[ln];
  } else {
    dst[ln] = make_float4(0.f, 0.f, 0.f, 0.f);
  }
  if (ln == 0) maskOut[(size_t)b * NTOK + p] = (p < K) ? 1.0f : 0.0f;
}

// -------------------------------------------------------------------------
extern "C" void kernel_launch(void* const* d_in, const int* in_sizes, int n_in,
                              void* d_out, int out_size, void* d_ws, size_t ws_size,
                              hipStream_t stream) {
  (void)in_sizes; (void)n_in; (void)out_size; (void)ws_size;
  const float* x = (const float*)d_in[0];   // (64,4096,256) f32
  const float* q = (const float*)d_in[1];   // (64,256)      f32

  float* feats   = (float*)d_out;                             // 64*4096*256
  float* maskOut = feats + (size_t)BATCH * NTOK * DIM;        // 64*4096

  float* y    = (float*)d_ws;                                 // 1 MB
  int*   perm = (int*)((char*)d_ws + (size_t)BATCH * NTOK * sizeof(float)); // 1 MB
  int*   Karr = perm + (size_t)BATCH * NTOK;                  // 256 B

  qvt_gemv_wmma<<<dim3(NTOK / 16, BATCH), 32, 0, stream>>>(x, q, y);
  qvt_tree<<<BATCH, 256, 0, stream>>>(y, perm, Karr);
  qvt_write<<<dim3(NTOK / 4, BATCH), 256, 0, stream>>>(x, perm, Karr, feats, maskOut);
}